// NearestNeighborDis_77309411647
// MI455X (gfx1250) — compile-verified
//
#include <hip/hip_runtime.h>

typedef float v2f __attribute__((ext_vector_type(2)));
typedef float v8f __attribute__((ext_vector_type(8)));

#define WAVES_PER_BLOCK 8
#define TILES_PER_WAVE  4
#define ROWS_PER_WAVE   (TILES_PER_WAVE * 16)             // 64
#define ROWS_PER_BLOCK  (WAVES_PER_BLOCK * ROWS_PER_WAVE) // 512
#define CHUNK_PTS       1024
#define N_POINTS        65536
#define M_POINTS        65536
#define NUM_BLOCKS      (N_POINTS / ROWS_PER_BLOCK)       // 128

// Single-instruction f32 min: bypasses OCML fminf (which drags in
// v_max_num(x,x) canonicalization) and select-lowering (which produced
// serialized v_cmp/v_cndmask chains through VCC).
static __device__ __forceinline__ float fmin_asm(float a, float b) {
    float r;
    asm("v_min_num_f32 %0, %1, %2" : "=v"(r) : "v"(a), "v"(b));
    return r;
}

__global__ __launch_bounds__(256)
void chamfer_main(const float* __restrict__ pc0,
                  const float* __restrict__ pc1,
                  float* __restrict__ blockRes /* [0..127]=sum, [128..255]=count */) {
    __shared__ float lds_b[CHUNK_PTS * 4];   // (-2x,-2y,-2z,b_sq) quads, 16KB

    const int tid  = threadIdx.x;
    const int lane = tid & 31;
    const int wave = tid >> 5;
    const int half = lane >> 4;   // 0: lanes 0-15, 1: lanes 16-31
    const int l16  = lane & 15;
    const int rowBase = blockIdx.x * ROWS_PER_BLOCK + wave * ROWS_PER_WAVE;

    // ---- A fragments: 16x4 f32, lanes0-15 hold K=0,1; lanes16-31 hold K=2,3
    // A-row = (x, y, z, 1)
    v2f afrag[TILES_PER_WAVE];
#pragma unroll
    for (int t = 0; t < TILES_PER_WAVE; ++t) {
        int row = rowBase + t * 16 + l16;
        const float* p = pc0 + row * 3;
        if (half == 0) { afrag[t][0] = p[0]; afrag[t][1] = p[1]; }
        else           { afrag[t][0] = p[2]; afrag[t][1] = 1.0f; }
    }

    v8f cmin[TILES_PER_WAVE];
#pragma unroll
    for (int t = 0; t < TILES_PER_WAVE; ++t)
#pragma unroll
        for (int k = 0; k < 8; ++k) cmin[t][k] = 3.0e38f;

    for (int m0 = 0; m0 < M_POINTS; m0 += CHUNK_PTS) {
        // ---- stage b' = (-2x,-2y,-2z, x^2+y^2+z^2) into LDS
#pragma unroll
        for (int i = 0; i < CHUNK_PTS / 256; ++i) {
            int pl = tid + i * 256;
            const float* q = pc1 + (size_t)(m0 + pl) * 3;
            float x = q[0], y = q[1], z = q[2];
            float* o = lds_b + pl * 4;
            o[0] = -2.0f * x;
            o[1] = -2.0f * y;
            o[2] = -2.0f * z;
            o[3] = x * x + y * y + z * z;
        }
        __syncthreads();

        // ---- sweep B tiles from LDS: B = 4x16 f32, lane layout symmetric to A.
        // Unroll 4: one DS base register + immediate offsets (stride 256B).
#pragma unroll 4
        for (int j = 0; j < CHUNK_PTS; j += 16) {
            const float* src = lds_b + (j + l16) * 4 + half * 2;
            v2f bfrag;
            bfrag[0] = src[0];
            bfrag[1] = src[1];
#pragma unroll
            for (int t = 0; t < TILES_PER_WAVE; ++t) {
                v8f c = {0.f, 0.f, 0.f, 0.f, 0.f, 0.f, 0.f, 0.f};
                // D[i][j] = a_i . b'_j = b_sq_j - 2 a_i.b_j   (exact fp32)
                v8f d = __builtin_amdgcn_wmma_f32_16x16x4_f32(
                            /*neg_a=*/false, afrag[t],
                            /*neg_b=*/false, bfrag,
                            /*c_mod=*/(short)0, c,
                            /*reuse_a=*/false, /*reuse_b=*/false);
#pragma unroll
                for (int k = 0; k < 8; ++k)
                    cmin[t][k] = fmin_asm(d[k], cmin[t][k]);
            }
        }
        __syncthreads();
    }

    // ---- reduce over the 16 columns held per half-wave (C/D layout:
    // VGPR k: lanes0-15 -> row k, lanes16-31 -> row k+8; column = lane%16)
#pragma unroll
    for (int t = 0; t < TILES_PER_WAVE; ++t)
#pragma unroll
        for (int k = 0; k < 8; ++k) {
            float v = cmin[t][k];
#pragma unroll
            for (int m = 8; m >= 1; m >>= 1)
                v = fmin_asm(v, __shfl_xor(v, m, 32));
            cmin[t][k] = v;
        }

    // lanes 0..7 own rows base+t*16+l16 ; lanes 16..23 own rows base+t*16+8+l16
    float lsum = 0.0f, lcnt = 0.0f;
    if (l16 < 8) {
#pragma unroll
        for (int t = 0; t < TILES_PER_WAVE; ++t) {
            float val = 0.0f;
#pragma unroll
            for (int k = 0; k < 8; ++k)
                if (k == l16) val = cmin[t][k];
            int row = rowBase + t * 16 + (half ? 8 + l16 : l16);
            const float* p = pc0 + row * 3;
            float asq = p[0] * p[0] + p[1] * p[1] + p[2] * p[2];
            float dist = val + asq;
            if (dist <= 2.0f) { lsum += dist; lcnt += 1.0f; }
        }
    }

    // ---- deterministic block-level reduction (fixed tree order), reuse LDS
    __syncthreads();
    float* red = lds_b;               // red[0..255]=sum, red[256..511]=count
    red[tid]       = lsum;
    red[256 + tid] = lcnt;
    __syncthreads();
#pragma unroll
    for (int s = 128; s > 0; s >>= 1) {
        if (tid < s) {
            red[tid]       += red[tid + s];
            red[256 + tid] += red[256 + tid + s];
        }
        __syncthreads();
    }
    if (tid == 0) {
        blockRes[blockIdx.x]              = red[0];
        blockRes[NUM_BLOCKS + blockIdx.x] = red[256];
    }
}

__global__ __launch_bounds__(NUM_BLOCKS)
void chamfer_finalize(const float* __restrict__ blockRes, float* __restrict__ out) {
    __shared__ float s[2 * NUM_BLOCKS];
    const int tid = threadIdx.x;
    s[tid]              = blockRes[tid];
    s[NUM_BLOCKS + tid] = blockRes[NUM_BLOCKS + tid];
    __syncthreads();
#pragma unroll
    for (int w = NUM_BLOCKS / 2; w > 0; w >>= 1) {
        if (tid < w) {
            s[tid]              += s[tid + w];
            s[NUM_BLOCKS + tid] += s[NUM_BLOCKS + tid + w];
        }
        __syncthreads();
    }
    if (tid == 0) out[0] = s[0] / s[NUM_BLOCKS];
}

extern "C" void kernel_launch(void* const* d_in, const int* in_sizes, int n_in,
                              void* d_out, int out_size, void* d_ws, size_t ws_size,
                              hipStream_t stream) {
    const float* pc0 = (const float*)d_in[0];
    const float* pc1 = (const float*)d_in[1];
    float* blockRes = (float*)d_ws;   // 2*NUM_BLOCKS floats
    float* out      = (float*)d_out;

    dim3 grid(NUM_BLOCKS);                  // 128 blocks
    dim3 block(WAVES_PER_BLOCK * 32);       // 256 threads = 8 waves
    chamfer_main<<<grid, block, 0, stream>>>(pc0, pc1, blockRes);

    chamfer_finalize<<<1, NUM_BLOCKS, 0, stream>>>(blockRes, out);
}